// ConeGeometryNet_14654428414496
// MI455X (gfx1250) — compile-verified
//
#include <hip/hip_runtime.h>

// ---------------------------------------------------------------------------
// ConeGeometryNet on MI455X (gfx1250, wave32)
//
// points (N,2) x 500 cones -> one N x 2000 f32 GEMM (K=2 padded to 4) via
// v_wmma_f32_16x16x4_f32, fused with containment/distance postprocessing.
// Coefficients staged block-wide into LDS with the Tensor Data Mover
// (tensor_load_to_lds + s_wait_tensorcnt). Cross-lane combines use
// v_permlane16_b32 (VALU, no DS waits). Outputs (~324 MB) are streamed with
// non-temporal stores -> kernel is store-bandwidth bound (~14 us @ 23.3 TB/s).
// ---------------------------------------------------------------------------

typedef float v2f __attribute__((ext_vector_type(2)));
typedef float v8f __attribute__((ext_vector_type(8)));
typedef int   v4i __attribute__((ext_vector_type(4)));
typedef int   v8i __attribute__((ext_vector_type(8)));

#define CC 20
#define HH 5
#define WW 4
#define NFRAME   100          // C*H
#define NWIN     400          // C*H*W
#define NCONES   500
#define NCOLS    2000         // 4 coefficient columns per cone
#define NTILES   125          // NCOLS / 16
#define FR_TILES 25           // first 25 tiles = frame cones, rest = windows

// Cross-lane xor within a 16-lane row: v_permlane16_b32 if available
// (pure VALU), else ds_swizzle_b32 with immediate xor-mask pattern.
template <unsigned SEL_LO, unsigned SEL_HI, int SWZ>
__device__ __forceinline__ float lane_xor(float x) {
#if __has_builtin(__builtin_amdgcn_permlane16)
    unsigned r = __builtin_amdgcn_permlane16(__float_as_uint(x), __float_as_uint(x),
                                             SEL_LO, SEL_HI, false, false);
    return __uint_as_float(r);
#else
    return __int_as_float(__builtin_amdgcn_ds_swizzle(__float_as_int(x), SWZ));
#endif
}
#define LANE_XOR1(x) (lane_xor<0x67452301u, 0xEFCDAB89u, 0x041F>(x))
#define LANE_XOR2(x) (lane_xor<0x54761032u, 0xDCFE98BAu, 0x081F>(x))
#define LANE_XOR4(x) (lane_xor<0x32107654u, 0xBA98FEDCu, 0x101F>(x))
#define LANE_XOR8(x) (lane_xor<0xFEDCBA98u, 0x76543210u, 0x201F>(x))

// ---------------------------------------------------------------------------
// Kernel 1: build the 2 x 2000 coefficient matrix (WMMA B operand).
// Column layout per cone i (j = 4*i):
//   j+0: (i00,i01) -> x0 ; j+1: (i10,i11) -> x1     (cone coordinates)
//   j+2: (r0x,r0y) -> d0 ; j+3: (r1x,r1y) -> d1     (|p x u| line distances)
// Row 0 multiplies px, row 1 multiplies py (WMMA K rows 2,3 are zero).
// ---------------------------------------------------------------------------
__global__ __launch_bounds__(512)
void cone_prep(const float* __restrict__ frame,
               const float* __restrict__ windows,
               float* __restrict__ Bp /* [2][NCOLS] */)
{
    int i = threadIdx.x;
    if (i >= NCONES) return;
    const float* src = (i < NFRAME) ? (frame + i * 4) : (windows + (i - NFRAME) * 4);
    float a = src[0], b = src[1];   // edge dir 0
    float c = src[2], d = src[3];   // edge dir 1
    float rdet = 1.0f / (a * d - b * c);      // det(dirs^T)
    float i00 =  d * rdet, i01 = -c * rdet;   // (dirs^T)^-1 rows
    float i10 = -b * rdet, i11 =  a * rdet;
    float rn0 = rsqrtf(a * a + b * b);
    float rn1 = rsqrtf(c * c + d * d);
    float r0x =  b * rn0, r0y = -a * rn0;     // coeffs of |p x u0|
    float r1x =  d * rn1, r1y = -c * rn1;     // coeffs of |p x u1|
    int j = i * 4;
    Bp[0 * NCOLS + j + 0] = i00;  Bp[1 * NCOLS + j + 0] = i01;
    Bp[0 * NCOLS + j + 1] = i10;  Bp[1 * NCOLS + j + 1] = i11;
    Bp[0 * NCOLS + j + 2] = r0x;  Bp[1 * NCOLS + j + 2] = r0y;
    Bp[0 * NCOLS + j + 3] = r1x;  Bp[1 * NCOLS + j + 3] = r1y;
}

// ---------------------------------------------------------------------------
// Kernel 2: fused GEMM + postprocess. One wave per 16-point M tile,
// 4 waves (64 points) per block.
// ---------------------------------------------------------------------------
__global__ __launch_bounds__(128)
void cone_main(const float* __restrict__ pts,
               const float* __restrict__ Bp,
               float* __restrict__ o_fc,  float* __restrict__ o_wc,
               float* __restrict__ o_hc,  float* __restrict__ o_ct,
               float* __restrict__ o_ifd, float* __restrict__ o_ofd,
               float* __restrict__ o_iwd, float* __restrict__ o_owd,
               int N)
{
    __shared__ float BpLDS[2 * NCOLS];         // 16 KB, TDM-staged coefficients
    __shared__ float wmaxLDS [4][16][NFRAME];  // per-wave: max_w window_inside
    __shared__ float houseLDS[4][16][NFRAME];  // per-wave: house containment

    const int lane   = threadIdx.x & 31;
    const int wave   = threadIdx.x >> 5;
    const int n_base = (blockIdx.x * 4 + wave) * 16;

    // ---- stage the 16 KB coefficient matrix into LDS with the TDM ---------
    if (wave == 0) {
        // D# group 0: count=1 | lds_addr | global_addr | type=2
        unsigned long long ga = (unsigned long long)(uintptr_t)Bp;
        unsigned int la = (unsigned int)(uintptr_t)&BpLDS[0]; // addr[31:0] = LDS offset
        v4i g0;
        g0[0] = 1;                                            // count=1, user D#
        g0[1] = (int)la;                                      // lds_addr
        g0[2] = (int)(ga & 0xffffffffu);                      // global_addr[31:0]
        g0[3] = (int)(((ga >> 32) & 0x1ffffffu) | 0x80000000u); // [56:32] | type=2
        // D# group 1: 1-D tensor of 4000 dwords, tile = whole tensor
        v8i g1;
        g1[0] = (2 << 16);                    // data_size = 4 bytes, no multicast
        g1[1] = (int)(4000u << 16);           // tensor_dim0[15:0] in bits 63:48
        g1[2] = (int)(1u << 16);              // dim0[31:16]=0 | tensor_dim1=1
        g1[3] = (int)(4000u << 16);           // dim1 hi=0 | tile_dim0=4000
        g1[4] = 1;                            // tile_dim1=1, tile_dim2=0
        g1[5] = 4000;                         // tensor_dim0_stride
        g1[6] = 0;
        g1[7] = 0;
        asm volatile("tensor_load_to_lds %0, %1" :: "s"(g0), "s"(g1) : "memory");
        __builtin_amdgcn_s_wait_tensorcnt(0);
    }
    __syncthreads();

    const int kh = lane >> 4;        // 0: A rows hold K=0/1 ; 1: K=2/3 (zero)
    const int t  = lane & 3;         // column role: x0,x1,d0,d1
    const int q  = (lane >> 2) & 3;  // cone within 16-column tile
    const bool lowt = (t < 2);

    // A operand: 16x4, lanes 0-15 = {px,py}, lanes 16-31 = K padding (zeros).
    int pidx = n_base + (lane & 15);
    pidx = pidx < N ? pidx : N - 1;
    float px = pts[pidx * 2 + 0];
    float py = pts[pidx * 2 + 1];
    v2f A;
    A.x = kh ? 0.0f : px;
    A.y = kh ? 0.0f : py;

    // ---- pass 1: window tiles (their max is needed by frame tiles) ---------
    for (int g = FR_TILES; g < NTILES; ++g) {
        v2f B;
        B.x = BpLDS[kh * NCOLS + g * 16 + (lane & 15)];
        B.y = 0.0f;
        v8f Dm = {};
        Dm = __builtin_amdgcn_wmma_f32_16x16x4_f32(
                 false, A, false, B, (short)0, Dm, false, false);

        const int gh = g - FR_TILES;  // (c*H + h); its 4 windows = this tile
#pragma unroll
        for (int r = 0; r < 8; ++r) {
            float v  = Dm[r];
            float pr = LANE_XOR1(v);
            float insC = (fminf(v, pr) >= 0.0f) ? 1.0f : 0.0f;   // x0>=0 && x1>=0
            float dstC = fminf(fabsf(v), fabsf(pr));             // min(|d0|,|d1|)
            float u  = lowt ? insC : dstC;
            float o  = LANE_XOR2(u);
            float ins = lowt ? u : o;
            float dst = lowt ? o : u;

            // max of window_inside over the 4 windows (q = w)
            float wm = fmaxf(ins, LANE_XOR4(ins));
            wm = fmaxf(wm, LANE_XOR8(wm));

            int prow = r + kh * 8;
            if ((lane & 15) == 3) wmaxLDS[wave][prow][gh] = wm;

            int n = n_base + prow;
            float  val = (t == 0) ? ins : ((t == 1) ? ins * dst : (1.0f - ins) * dst);
            float* ptr = (t == 0) ? o_wc : ((t == 1) ? o_iwd : o_owd);
            if (t < 3 && n < N)
                __builtin_nontemporal_store(val, ptr + (long)n * NWIN + gh * 4 + q);
        }
    }

    // ---- pass 2: frame tiles ----------------------------------------------
    for (int g = 0; g < FR_TILES; ++g) {
        v2f B;
        B.x = BpLDS[kh * NCOLS + g * 16 + (lane & 15)];
        B.y = 0.0f;
        v8f Dm = {};
        Dm = __builtin_amdgcn_wmma_f32_16x16x4_f32(
                 false, A, false, B, (short)0, Dm, false, false);

#pragma unroll
        for (int r = 0; r < 8; ++r) {
            float v  = Dm[r];
            float pr = LANE_XOR1(v);
            float insC = (fminf(v, pr) >= 0.0f) ? 1.0f : 0.0f;
            float dstC = fminf(fabsf(v), fabsf(pr));
            float u  = lowt ? insC : dstC;
            float o  = LANE_XOR2(u);
            float ins = lowt ? u : o;
            float dst = lowt ? o : u;

            int prow = r + kh * 8;
            int f = g * 4 + q;                        // frame cone index c*H + h
            float wm    = wmaxLDS[wave][prow][f];
            float house = ins * (1.0f - wm);
            if (t == 0) houseLDS[wave][prow][f] = house;

            int n = n_base + prow;
            float  val = (t == 0) ? ins
                       : ((t == 1) ? ins * dst
                       : ((t == 2) ? (1.0f - ins) * dst : house));
            float* ptr = (t == 0) ? o_fc
                       : ((t == 1) ? o_ifd
                       : ((t == 2) ? o_ofd : o_hc));
            if (n < N)
                __builtin_nontemporal_store(val, ptr + (long)n * NFRAME + f);
        }
    }

    // ---- pass 3: containment = max over h of house -------------------------
    for (int it = 0; it < 10; ++it) {
        int idx  = it * 32 + lane;        // 0..319 = 16 points x 20 classes
        int prow = idx / CC;
        int c    = idx % CC;
        float m = houseLDS[wave][prow][c * HH + 0];
        m = fmaxf(m, houseLDS[wave][prow][c * HH + 1]);
        m = fmaxf(m, houseLDS[wave][prow][c * HH + 2]);
        m = fmaxf(m, houseLDS[wave][prow][c * HH + 3]);
        m = fmaxf(m, houseLDS[wave][prow][c * HH + 4]);
        int n = n_base + prow;
        if (n < N)
            __builtin_nontemporal_store(m, o_ct + (long)n * CC + c);
    }
}

// ---------------------------------------------------------------------------
extern "C" void kernel_launch(void* const* d_in, const int* in_sizes, int n_in,
                              void* d_out, int out_size, void* d_ws, size_t ws_size,
                              hipStream_t stream) {
    const float* pts   = (const float*)d_in[0];  // (N, 2)
    const float* frame = (const float*)d_in[1];  // (C, H, 2, 2)
    const float* win   = (const float*)d_in[2];  // (C, H, W, 2, 2)
    const int N = in_sizes[0] / 2;

    float* Bp  = (float*)d_ws;                   // 2 x 2000 coefficients (16 KB)
    float* out = (float*)d_out;

    // Output layout: dict order, all flattened float32.
    long off = 0;
    float* o_emb = out + off; off += (long)N * 2;       // embeddings
    float* o_fc  = out + off; off += (long)N * NFRAME;  // crisp_frame_containment
    float* o_wc  = out + off; off += (long)N * NWIN;    // crisp_window_containment
    float* o_hc  = out + off; off += (long)N * NFRAME;  // crisp_house_containment
    float* o_ct  = out + off; off += (long)N * CC;      // crisp_containment
    float* o_ifd = out + off; off += (long)N * NFRAME;  // inner_frame_distance
    float* o_ofd = out + off; off += (long)N * NFRAME;  // outer_frame_distance
    float* o_iwd = out + off; off += (long)N * NWIN;    // inner_window_distances
    float* o_owd = out + off; off += (long)N * NWIN;    // outer_window_distances

    // embeddings is just a copy of points
    hipMemcpyAsync(o_emb, pts, sizeof(float) * 2 * N, hipMemcpyDeviceToDevice, stream);

    cone_prep<<<1, 512, 0, stream>>>(frame, win, Bp);

    const int nblk = (N + 63) / 64;  // 64 points per block (4 waves x 16)
    cone_main<<<nblk, 128, 0, stream>>>(pts, Bp,
                                        o_fc, o_wc, o_hc, o_ct,
                                        o_ifd, o_ofd, o_iwd, o_owd, N);
}